// HashDistill_26980984554133
// MI455X (gfx1250) — compile-verified
//
#include <hip/hip_runtime.h>

typedef __attribute__((ext_vector_type(2))) float v2f;
typedef __attribute__((ext_vector_type(8))) float v8f;

#define D_DIM   256
#define NUM_K   8192
#define CCOLS   64          // columns per streamed chunk
#define CSTRIDE 72          // 64 + 8 pad floats: 288B row stride (16B aligned), 8-bank skew
#define NCHUNK  4           // 256 / 64

// ---- CDNA5 async direct-to-LDS DMA (GLOBAL_LOAD_ASYNC_TO_LDS_B128, tracked by ASYNCcnt).
__device__ __forceinline__ void async_ld_b128(unsigned lds_off, const float* g) {
    asm volatile("global_load_async_to_lds_b128 %0, %1, off"
                 :: "v"(lds_off), "v"(g) : "memory");
}
template <int N>
__device__ __forceinline__ void wait_asynccnt() {
    asm volatile("s_wait_asynccnt %0" :: "n"(N) : "memory");
}
__device__ __forceinline__ void wait_dscnt0() {
    // Async DMA writes to LDS are NOT ordered with this wave's DS ops; drain DS
    // before re-issuing DMA into a just-processed buffer.
    asm volatile("s_wait_dscnt 0" ::: "memory");
}

// Issue one 32x64 chunk (8 KB) as 16 async b128 ops: each op moves 2 x 256 contiguous
// bytes (lanes 0-15 -> row 2t segment, lanes 16-31 -> row 2t+1 segment).
__device__ __forceinline__ void issue_chunk(const float* __restrict__ sS,
                                            const float* __restrict__ sT,
                                            int q, float* b, int lane) {
    const int sub = lane >> 4;          // 0 or 1: which of the two rows this op covers
    const int cb  = (lane & 15) * 4;    // float offset inside the 64-col chunk (16B units)
#pragma unroll
    for (int t = 0; t < 16; ++t) {
        const int r  = 2 * t + sub;     // tile row 0..31 (0..15 = xS, 16..31 = xT)
        const int rl = r & 15;
        const float* base = (t < 8) ? sS : sT;
        const float* g = base + (size_t)rl * D_DIM + q * CCOLS + cb;
        unsigned lds = (unsigned)(size_t)(b + r * CSTRIDE + cb); // low 32b of flat ptr = LDS byte addr
        async_ld_b128(lds, g);
    }
}

// Per-chunk: column means (over all 32 rows), mix toward mean in place, then
// accumulate the UNNORMALIZED gram with V_WMMA_F32_16X16X4_F32.
__device__ __forceinline__ void process_chunk(float* b, int lane, int arow, int koff,
                                              const float* pmld,
                                              v8f& c00, v8f& c01, v8f& c11) {
    // Lane owns columns {lane, lane+32}: consecutive banks, conflict-free.
    float xx0 = 0.f, xx1 = 0.f;
#pragma unroll
    for (int r = 0; r < 32; ++r) {
        xx0 += b[r * CSTRIDE + lane];
        xx1 += b[r * CSTRIDE + lane + 32];
    }
    xx0 *= (1.0f / 32.0f);              // == 0.5*(mean_s + mean_t)
    xx1 *= (1.0f / 32.0f);
#pragma unroll
    for (int r = 0; r < 32; ++r) {
        float m  = pmld[r];
        float om = 1.0f - m;
        float* p = b + r * CSTRIDE + lane;
        p[0]  = m * p[0]  + om * xx0;
        p[32] = m * p[32] + om * xx1;
    }
    // A-operand layout for f32 16x16x4: lanes 0-15 (M=lane, K=0/1), lanes 16-31 (K=2/3).
    // For the gram (B = A^T) the B operand of column-block J equals the A operand of
    // row-block J, so one ds_load_b64 per row-block feeds all three tiles.
#pragma unroll
    for (int kk = 0; kk < 16; ++kk) {
        int col = 4 * kk + koff;        // even -> 8B aligned
        v2f p0 = *(const v2f*)(b + arow * CSTRIDE + col);          // rows 0..15
        v2f p1 = *(const v2f*)(b + (arow + 16) * CSTRIDE + col);   // rows 16..31
        c00 = __builtin_amdgcn_wmma_f32_16x16x4_f32(false, p0, false, p0, (short)0, c00, false, false);
        c01 = __builtin_amdgcn_wmma_f32_16x16x4_f32(false, p0, false, p1, (short)0, c01, false, false);
        c11 = __builtin_amdgcn_wmma_f32_16x16x4_f32(false, p1, false, p1, (short)0, c11, false, false);
    }
}

// One wave (32 lanes) per group k; 64-column double-buffered async pipeline.
__global__ __launch_bounds__(32) void hash_distill_group_kernel(
    const float* __restrict__ xS, const float* __restrict__ xT,
    const float* __restrict__ pos_mask, float* __restrict__ gsum)
{
    __shared__ __align__(16) float buf[2][32 * CSTRIDE];   // 2 x 9216 B
    __shared__ float pmld[32];
    __shared__ float diag[32];
    __shared__ float invl[32];

    const int k    = blockIdx.x;
    const int lane = threadIdx.x;
    const float* sS = xS + (size_t)k * 16 * D_DIM;
    const float* sT = xT + (size_t)k * 16 * D_DIM;

    pmld[lane] = pos_mask[(size_t)k * 32 + lane];   // rows 0..15 = ms, 16..31 = mt

    issue_chunk(sS, sT, 0, &buf[0][0], lane);
    issue_chunk(sS, sT, 1, &buf[1][0], lane);

    v8f c00 = {}, c01 = {}, c11 = {};
    const int arow = lane & 15;
    const int koff = (lane >> 4) << 1;  // 0 for lanes 0-15, 2 for lanes 16-31

    wait_asynccnt<16>();                                    // chunk 0 landed
    process_chunk(&buf[0][0], lane, arow, koff, pmld, c00, c01, c11);
    wait_dscnt0();
    issue_chunk(sS, sT, 2, &buf[0][0], lane);

    wait_asynccnt<16>();                                    // chunk 1 landed
    process_chunk(&buf[1][0], lane, arow, koff, pmld, c00, c01, c11);
    wait_dscnt0();
    issue_chunk(sS, sT, 3, &buf[1][0], lane);

    wait_asynccnt<16>();                                    // chunk 2 landed
    process_chunk(&buf[0][0], lane, arow, koff, pmld, c00, c01, c11);

    wait_asynccnt<0>();                                     // chunk 3 landed
    process_chunk(&buf[1][0], lane, arow, koff, pmld, c00, c01, c11);

    // ---- Row norms from the gram diagonal: G_rr = sum_c v_rc^2.
    // C layout: lanes 0-15 (M=vgpr, N=lane), lanes 16-31 (M=vgpr+8, N=lane-16).
    const int ncol  = lane & 15;
    const int mbase = (lane >> 4) * 8;
#pragma unroll
    for (int v = 0; v < 8; ++v) {
        int M = mbase + v;
        if (ncol == M) {                 // this lane holds the diagonal entry of row M
            diag[M]      = c00[v];       // rows 0..15
            diag[16 + M] = c11[v];       // rows 16..31
        }
    }
    __syncthreads();
    invl[lane] = 1.0f / sqrtf(diag[lane]);
    __syncthreads();

    // ---- Strict-upper-triangle of normalized sim = G_ij * inv_i * inv_j:
    // tile00 strict-upper + tile11 strict-upper + ALL of tile01 (rows 0-15 x cols 16-31).
    const float iC0 = invl[ncol];
    const float iC1 = invl[16 + ncol];
    float part = 0.f;
#pragma unroll
    for (int v = 0; v < 8; ++v) {
        int M = mbase + v;
        float iM0 = invl[M];
        float iM1 = invl[16 + M];
        part += c01[v] * iM0 * iC1;
        if (ncol > M) part += c00[v] * iM0 * iC0 + c11[v] * iM1 * iC1;
    }
#pragma unroll
    for (int off = 16; off > 0; off >>= 1) part += __shfl_xor(part, off, 32);
    if (lane == 0) gsum[k] = part;
}

// Deterministic final reduction of 8192 per-group sums -> scalar.
__global__ __launch_bounds__(256) void hash_distill_final_reduce(
    const float* __restrict__ gsum, float* __restrict__ out)
{
    __shared__ float sm[256];
    int t = threadIdx.x;
    float s = 0.f;
    for (int i = t; i < NUM_K; i += 256) s += gsum[i];
    sm[t] = s;
    __syncthreads();
    for (int off = 128; off > 0; off >>= 1) {
        if (t < off) sm[t] += sm[t + off];
        __syncthreads();
    }
    if (t == 0) {
        // per_group_mean = sum/496 ; hkd = sum_k(...)/131072 ; return -hkd
        out[0] = -sm[0] * (1.0f / (496.0f * 131072.0f));
    }
}

extern "C" void kernel_launch(void* const* d_in, const int* in_sizes, int n_in,
                              void* d_out, int out_size, void* d_ws, size_t ws_size,
                              hipStream_t stream) {
    (void)in_sizes; (void)n_in; (void)out_size; (void)ws_size;
    const float* xS       = (const float*)d_in[0];
    const float* xT       = (const float*)d_in[1];
    const float* pos_mask = (const float*)d_in[2];
    float* gsum = (float*)d_ws;            // NUM_K floats of scratch
    float* out  = (float*)d_out;

    hash_distill_group_kernel<<<NUM_K, 32, 0, stream>>>(xS, xT, pos_mask, gsum);
    hash_distill_final_reduce<<<1, 256, 0, stream>>>(gsum, out);
}